// SemanticHeadSelectionAttention_79903571574881
// MI455X (gfx1250) — compile-verified
//
#include <hip/hip_runtime.h>
#include <hip/hip_bf16.h>
#include <math.h>

typedef __attribute__((ext_vector_type(2))) float v2f;
typedef __attribute__((ext_vector_type(8))) float v8f;

#define DIM   768
#define NH    12
#define HD    64
#define BATCH 4
#define SEQ   1024
#define H_ON  6
#define EPSM  0.1f

// ---------------------------------------------------------------------------
// Kernel 0: zero the per-head score accumulators
// ---------------------------------------------------------------------------
__global__ void shsa_init_kernel(float* __restrict__ scores) {
    if (threadIdx.x < NH) scores[threadIdx.x] = 0.0f;
}

// ---------------------------------------------------------------------------
// Kernel 1: QKV projection.  out = x @ W^T + b, stored as [B,H,N,D].
// One wave computes a 64(token) x 16(outfeat) super-tile for one of Q/K/V,
// reusing the weight B-fragment across 4 WMMA accumulators.
// A-matrix (16x4 f32) lane layout: lane l holds rows M=l%16, K = j + 2*(l/16).
// B-matrix (4x16 f32): lane l holds cols N=l%16, K = j + 2*(l/16).
// ---------------------------------------------------------------------------
__global__ __launch_bounds__(256)
void shsa_qkv_proj_kernel(const float* __restrict__ x,
                          const float* __restrict__ Wq, const float* __restrict__ bq,
                          const float* __restrict__ Wk, const float* __restrict__ bk,
                          const float* __restrict__ Wv, const float* __restrict__ bv,
                          float* __restrict__ q, float* __restrict__ k,
                          float* __restrict__ v) {
    const int lane = threadIdx.x & 31;
    const int wid  = blockIdx.x * 8 + (threadIdx.x >> 5);   // 0 .. 9215
    const int mat  = wid / (64 * 48);                       // 0=Q 1=K 2=V
    const int rem  = wid % (64 * 48);
    const int m0   = (rem / 48) * 64;                       // token super-tile
    const int o0   = (rem % 48) * 16;                       // out-feature tile

    const float* W    = (mat == 0) ? Wq : (mat == 1) ? Wk : Wv;
    const float* bias = (mat == 0) ? bq : (mat == 1) ? bk : bv;
    float*       out  = (mat == 0) ? q  : (mat == 1) ? k  : v;

    const int lo  = lane & 15;           // M (A) / N (B) sub-index
    const int hi2 = (lane >> 4) << 1;    // K sub-offset: 0 or 2

    const float* wrow = W + (o0 + lo) * DIM + hi2;
    const float* xr0  = x + (m0 +  0 + lo) * DIM + hi2;
    const float* xr1  = x + (m0 + 16 + lo) * DIM + hi2;
    const float* xr2  = x + (m0 + 32 + lo) * DIM + hi2;
    const float* xr3  = x + (m0 + 48 + lo) * DIM + hi2;

    v8f acc0 = {}, acc1 = {}, acc2 = {}, acc3 = {};
    for (int k0 = 0; k0 < DIM; k0 += 4) {
        v2f bf = *(const v2f*)(wrow + k0);
        v2f a0 = *(const v2f*)(xr0 + k0);
        v2f a1 = *(const v2f*)(xr1 + k0);
        v2f a2 = *(const v2f*)(xr2 + k0);
        v2f a3 = *(const v2f*)(xr3 + k0);
        acc0 = __builtin_amdgcn_wmma_f32_16x16x4_f32(false, a0, false, bf, (short)0, acc0, false, false);
        acc1 = __builtin_amdgcn_wmma_f32_16x16x4_f32(false, a1, false, bf, (short)0, acc1, false, false);
        acc2 = __builtin_amdgcn_wmma_f32_16x16x4_f32(false, a2, false, bf, (short)0, acc2, false, false);
        acc3 = __builtin_amdgcn_wmma_f32_16x16x4_f32(false, a3, false, bf, (short)0, acc3, false, false);
    }

    const float bval = bias[o0 + lo];
    const int   o    = o0 + lo;
    const int   h    = o >> 6;
    const int   d    = o & 63;
    const int   mb   = (lane >> 4) << 3;   // M offset 0 or 8 within tile

    #pragma unroll
    for (int t = 0; t < 4; ++t) {
        v8f a = (t == 0) ? acc0 : (t == 1) ? acc1 : (t == 2) ? acc2 : acc3;
        #pragma unroll
        for (int r = 0; r < 8; ++r) {
            int token = m0 + t * 16 + r + mb;       // global token index
            int b_    = token >> 10;
            int n     = token & (SEQ - 1);
            out[(((b_ * NH + h) << 10) | n) * HD + d] = a[r] + bval;
        }
    }
}

// ---------------------------------------------------------------------------
// Kernel 2: attention for one (b,h,16-row) tile per wave.
// LDS holds a 16 x 1024 fp32 logits strip (64 KB dynamic shared).
// ---------------------------------------------------------------------------
__global__ __launch_bounds__(32)
void shsa_attn_kernel(const float* __restrict__ q, const float* __restrict__ k,
                      const float* __restrict__ v, float* __restrict__ o,
                      float* __restrict__ scores) {
    extern __shared__ float s_p[];       // 16 * SEQ floats
    const int lane  = threadIdx.x;
    const int mtile = blockIdx.x & 63;
    const int bh    = blockIdx.x >> 6;   // 0..47
    const int h     = bh % NH;
    const int m0    = mtile * 16;

    const float* Q = q + (long)bh * SEQ * HD;
    const float* K = k + (long)bh * SEQ * HD;
    const float* V = v + (long)bh * SEQ * HD;
    float*       O = o + (long)bh * SEQ * HD;

    const int lo  = lane & 15;
    const int hi2 = (lane >> 4) << 1;
    const int mb  = (lane >> 4) << 3;

    // preload 16 Q A-fragments covering K-dim 0..63
    v2f aq[16];
    const float* qrow = Q + (m0 + lo) * HD + hi2;
    #pragma unroll
    for (int f = 0; f < 16; ++f) aq[f] = *(const v2f*)(qrow + 4 * f);

    // ---- phase 1: logits = Q K^T / sqrt(D) into LDS ----
    for (int nt = 0; nt < SEQ / 16; ++nt) {
        v8f acc = {};
        const float* krow = K + (nt * 16 + lo) * HD + hi2;
        #pragma unroll
        for (int f = 0; f < 16; ++f) {
            v2f bf = *(const v2f*)(krow + 4 * f);
            acc = __builtin_amdgcn_wmma_f32_16x16x4_f32(false, aq[f], false, bf, (short)0, acc, false, false);
        }
        #pragma unroll
        for (int r = 0; r < 8; ++r)
            s_p[(mb + r) * SEQ + nt * 16 + lo] = acc[r] * 0.125f;
    }
    __syncthreads();

    // ---- phase 2: softmax (lane handles half of row `lo`) ----
    float* base = s_p + lo * SEQ + (lane >> 4) * (SEQ / 2);
    float  mx   = -3.4e38f;
    for (int i = 0; i < SEQ / 2; i += 4) {
        float4 t = *(const float4*)(base + i);
        mx = fmaxf(mx, fmaxf(fmaxf(t.x, t.y), fmaxf(t.z, t.w)));
    }
    mx = fmaxf(mx, __shfl_xor(mx, 16, 32));

    float zs = 0.0f;
    for (int i = 0; i < SEQ / 2; i += 4) {
        float4 t = *(float4*)(base + i);
        t.x = __expf(t.x - mx); t.y = __expf(t.y - mx);
        t.z = __expf(t.z - mx); t.w = __expf(t.w - mx);
        *(float4*)(base + i) = t;
        zs += t.x + t.y + t.z + t.w;
    }
    zs += __shfl_xor(zs, 16, 32);
    const float rz = 1.0f / zs;

    // per-head importance: sum of normalized attention (== ~1 per row)
    float rowsum = zs * rz;
    float tot = rowsum;
    for (int s = 1; s < 32; s <<= 1) tot += __shfl_xor(tot, s, 32);
    tot *= 0.5f;                       // each row counted by two lanes
    if (lane == 0) atomicAdd(&scores[h], tot);
    __syncthreads();

    // ---- phase 3: O = P V  (P read from LDS, scaled by 1/Z on the fly) ----
    v8f av0 = {}, av1 = {}, av2 = {}, av3 = {};
    for (int k0 = 0; k0 < SEQ; k0 += 4) {
        v2f af = *(const v2f*)(s_p + lo * SEQ + k0 + hi2);
        af[0] *= rz; af[1] *= rz;      // lane's A-rows are row `lo`
        const float* vb = V + (k0 + hi2) * HD;
        v2f b0; b0[0] = vb[ 0 + lo];      b0[1] = vb[HD +  0 + lo];
        v2f b1; b1[0] = vb[16 + lo];      b1[1] = vb[HD + 16 + lo];
        v2f b2; b2[0] = vb[32 + lo];      b2[1] = vb[HD + 32 + lo];
        v2f b3; b3[0] = vb[48 + lo];      b3[1] = vb[HD + 48 + lo];
        av0 = __builtin_amdgcn_wmma_f32_16x16x4_f32(false, af, false, b0, (short)0, av0, false, false);
        av1 = __builtin_amdgcn_wmma_f32_16x16x4_f32(false, af, false, b1, (short)0, av1, false, false);
        av2 = __builtin_amdgcn_wmma_f32_16x16x4_f32(false, af, false, b2, (short)0, av2, false, false);
        av3 = __builtin_amdgcn_wmma_f32_16x16x4_f32(false, af, false, b3, (short)0, av3, false, false);
    }
    #pragma unroll
    for (int t = 0; t < 4; ++t) {
        v8f a = (t == 0) ? av0 : (t == 1) ? av1 : (t == 2) ? av2 : av3;
        #pragma unroll
        for (int r = 0; r < 8; ++r)
            O[(m0 + mb + r) * HD + t * 16 + lo] = a[r];
    }
}

// ---------------------------------------------------------------------------
// Kernel 3: top-H_ON head selection (stable tie-break like jax.lax.top_k)
// ---------------------------------------------------------------------------
__global__ void shsa_select_kernel(const float* __restrict__ scores,
                                   float* __restrict__ scale) {
    int h = threadIdx.x;
    if (h < NH) {
        float sh = scores[h];
        int rank = 0;
        for (int j = 0; j < NH; ++j) {
            float sj = scores[j];
            rank += (sj > sh) || (sj == sh && j < h);
        }
        const float on_val = ((float)NH - (float)(NH - H_ON) * EPSM) / (float)H_ON;
        scale[h] = (rank < H_ON) ? on_val : EPSM;
    }
}

// ---------------------------------------------------------------------------
// Kernel 4: apply head scale and relayout [B,H,N,D] -> [B,N,H*D]
// ---------------------------------------------------------------------------
__global__ __launch_bounds__(256)
void shsa_finalize_kernel(const float* __restrict__ o,
                          const float* __restrict__ scale,
                          float* __restrict__ out) {
    int g = blockIdx.x * blockDim.x + threadIdx.x;
    if (g >= BATCH * SEQ * DIM / 4) return;
    int flat   = g * 4;
    int within = flat % DIM;
    int h      = within / HD;
    int d      = within % HD;
    int n      = (flat / DIM) & (SEQ - 1);
    int b      = flat / (DIM * SEQ);
    float4 src = *(const float4*)(o + (((long)(b * NH + h) * SEQ + n) * HD + d));
    float  s   = scale[h];
    float4 r;
    r.x = src.x * s; r.y = src.y * s; r.z = src.z * s; r.w = src.w * s;
    *(float4*)(out + flat) = r;
}

// ---------------------------------------------------------------------------
extern "C" void kernel_launch(void* const* d_in, const int* in_sizes, int n_in,
                              void* d_out, int out_size, void* d_ws, size_t ws_size,
                              hipStream_t stream) {
    (void)in_sizes; (void)n_in; (void)out_size; (void)ws_size;
    const float* x  = (const float*)d_in[0];
    const float* Wq = (const float*)d_in[1];
    const float* bq = (const float*)d_in[2];
    const float* Wk = (const float*)d_in[3];
    const float* bk = (const float*)d_in[4];
    const float* Wv = (const float*)d_in[5];
    const float* bv = (const float*)d_in[6];
    // d_in[7] = num_visual_tokens (unused by the reference computation)

    float* ws = (float*)d_ws;
    const long SZ = (long)BATCH * NH * SEQ * HD;   // 3,145,728 floats
    float* q      = ws;
    float* k      = ws + SZ;
    float* v      = ws + 2 * SZ;
    float* o      = ws + 3 * SZ;
    float* scores = ws + 4 * SZ;
    float* scale  = scores + 16;

    shsa_init_kernel<<<1, 32, 0, stream>>>(scores);
    shsa_qkv_proj_kernel<<<1152, 256, 0, stream>>>(x, Wq, bq, Wk, bk, Wv, bv, q, k, v);
    shsa_attn_kernel<<<BATCH * NH * (SEQ / 16), 32, 16 * SEQ * sizeof(float), stream>>>(
        q, k, v, o, scores);
    shsa_select_kernel<<<1, 32, 0, stream>>>(scores, scale);
    shsa_finalize_kernel<<<BATCH * SEQ * DIM / 4 / 256, 256, 0, stream>>>(
        o, scale, (float*)d_out);
}